// CRF_25726854103571
// MI455X (gfx1250) — compile-verified
//
#include <hip/hip_runtime.h>

#define BB 64
#define TT 512
#define DD 1024
#define KK 50
#define MROWS (BB * TT)   // 32768

typedef __attribute__((ext_vector_type(2))) float v2f;
typedef __attribute__((ext_vector_type(8))) float v8f;

// ---------------------------------------------------------------------------
// Kernel 1: emissions GEMM  scores[r, n] = (sum_d X[r,d] * W[d,n] + b[n]) * mask[r]
// One wave per 16x16 output tile, V_WMMA_F32_16X16X4_F32 over D in steps of 4.
// Block = 128 threads (4 waves); wave w handles N-tile w (n0 = 16*w), sharing
// the A tile through L1.
//
// Columns n >= 50 are padding: WMMA output (M,N) depends only on A-row M and
// B-column N, so padded columns need only *in-bounds* addresses (clamped),
// not zeros — their accumulator lanes are garbage but never stored. This
// keeps the inner loop branch-free (no exec-mask save/restore per step).
// ---------------------------------------------------------------------------
__global__ __launch_bounds__(128) void crf_emissions_gemm(
    const float* __restrict__ X,     // [MROWS, DD]
    const float* __restrict__ W,     // [DD, KK]
    const float* __restrict__ bias,  // [KK]
    const int*   __restrict__ mask,  // [MROWS]
    float* __restrict__ out)         // [MROWS, KK]
{
  const int lane  = threadIdx.x & 31;
  const int wave  = threadIdx.x >> 5;   // N-tile index 0..3
  const int r0    = blockIdx.x * 16;    // M-tile base row
  const int half  = lane >> 4;          // 0: K=d,d+1   1: K=d+2,d+3
  const int l16   = lane & 15;

  const int n      = wave * 16 + l16;               // output column
  const int n_safe = (n < KK) ? n : (KK - 1);       // clamped for loads

  v8f acc = {};
  // A pointer: row r0+l16, starting at K-offset 2*half
  const float* xp = X + (size_t)(r0 + l16) * DD + half * 2;
  // B pointer: W[(2*half)*KK + n_safe]; per step bv = {wp[0], wp[KK]}, wp += 4*KK
  const float* wp = W + (size_t)(half * 2) * KK + n_safe;

#pragma unroll 4
  for (int d = 0; d < DD; d += 4) {
    v2f a = *(const v2f*)xp;            // global_load_b64
    v2f bv;
    bv.x = wp[0];                        // global_load_b32
    bv.y = wp[KK];                       // global_load_b32 offset:200
    acc = __builtin_amdgcn_wmma_f32_16x16x4_f32(
        /*neg_a=*/false, a, /*neg_b=*/false, bv,
        /*c_mod=*/(short)0, acc, /*reuse_a=*/false, /*reuse_b=*/false);
    xp += 4;
    wp += 4 * KK;
  }

  if (n < KK) {
    const float bb = bias[n];
#pragma unroll
    for (int i = 0; i < 8; ++i) {
      // C/D layout: VGPR i holds M = i (lanes 0-15) / M = i+8 (lanes 16-31)
      const int row = r0 + i + half * 8;
      const float mf = (float)mask[row];
      out[(size_t)row * KK + n] = (acc[i] + bb) * mf;
    }
  }
}

// ---------------------------------------------------------------------------
// Kernel 2: per-batch Viterbi + log-partition + gold score.
// One block (64 threads, 2 waves) per batch; state j per thread.
// Backpointers for all 512 steps kept in LDS (32KB of the 320KB WGP LDS).
// ---------------------------------------------------------------------------
__global__ __launch_bounds__(64) void crf_dp(
    const float* __restrict__ scores,  // [BB, TT, KK] (already masked)
    const int*   __restrict__ mask,    // [BB, TT]
    const int*   __restrict__ labels,  // [BB, TT]
    const float* __restrict__ trans,   // [KK, KK]
    const float* __restrict__ startv,  // [KK]
    const float* __restrict__ endv,    // [KK]
    float* __restrict__ tags_out,      // [BB, TT] stored as float
    float* __restrict__ ll_ws)         // [BB]
{
  __shared__ float s_trans[KK * KK];
  __shared__ float aV[64];
  __shared__ float aP[64];
  __shared__ float red[64];
  __shared__ float s_logZ;
  __shared__ unsigned char bp[TT * 64];   // bp[t][j], t >= 1 used

  const int b = blockIdx.x;
  const int j = threadIdx.x;
  const float NEG = -3.0e38f;

  for (int idx = j; idx < KK * KK; idx += 64) s_trans[idx] = trans[idx];

  const float* sc   = scores + (size_t)b * TT * KK;
  const int*   mrow = mask   + b * TT;
  const int*   lrow = labels + b * TT;

  if (j < KK) {
    const float a0 = startv[j] + sc[j];
    aV[j] = a0;
    aP[j] = a0;
  } else {
    aV[j] = NEG;
    aP[j] = NEG;
  }
  __syncthreads();

  for (int t = 1; t < TT; ++t) {
    float newV = 0.0f, newP = 0.0f;
    int   arg  = j;
    const int m = mrow[t];

    if (j < KK) {
      const float emit = sc[(size_t)t * KK + j];
      float vmax = NEG, pmax = NEG;
      int   vidx = 0;
      for (int i = 0; i < KK; ++i) {
        const float tr = s_trans[i * KK + j];
        const float xv = aV[i] + tr;
        if (xv > vmax) { vmax = xv; vidx = i; }   // first-max, matches jnp.argmax
        const float xp = aP[i] + tr;
        pmax = fmaxf(pmax, xp);
      }
      float s = 0.0f;
      for (int i = 0; i < KK; ++i) {
        s += __expf(aP[i] + s_trans[i * KK + j] - pmax);
      }
      newV = vmax + emit;
      newP = pmax + __logf(s) + emit;
      arg  = vidx;
    }
    __syncthreads();
    if (j < KK) {
      if (m > 0) {
        aV[j] = newV;
        aP[j] = newP;
        bp[t * 64 + j] = (unsigned char)arg;
      } else {
        bp[t * 64 + j] = (unsigned char)j;  // identity backpointer when masked
      }
    }
    __syncthreads();
  }

  // ---- Viterbi termination + backtrack (thread 0, deterministic) ----
  red[j] = (j < KK) ? (aV[j] + endv[j]) : NEG;
  __syncthreads();
  if (j == 0) {
    float best = red[0];
    int   last = 0;
    for (int i = 1; i < KK; ++i)
      if (red[i] > best) { best = red[i]; last = i; }
    int tag = last;
    for (int t = TT - 1; t >= 1; --t) {
      tags_out[b * TT + t] = (mrow[t] > 0) ? (float)tag : 0.0f;
      tag = (int)bp[t * 64 + tag];
    }
    tags_out[b * TT + 0] = (mrow[0] > 0) ? (float)tag : 0.0f;
  }
  __syncthreads();

  // ---- log partition ----
  red[j] = (j < KK) ? (aP[j] + endv[j]) : NEG;
  __syncthreads();
  if (j == 0) {
    float m = red[0];
    for (int i = 1; i < KK; ++i) m = fmaxf(m, red[i]);
    float s = 0.0f;
    for (int i = 0; i < KK; ++i) s += __expf(red[i] - m);
    s_logZ = m + __logf(s);
  }
  __syncthreads();

  // ---- gold score (strided partial sums) ----
  float part = 0.0f;
  for (int t = j; t < TT; t += 64) {
    const float mf  = (float)mrow[t];
    const int   lab = lrow[t];
    part += sc[(size_t)t * KK + lab] * mf;
    if (t >= 1) part += s_trans[lrow[t - 1] * KK + lab] * mf;
  }
  red[j] = part;
  __syncthreads();
  if (j == 0) {
    float gsum = 0.0f;
    for (int i = 0; i < 64; ++i) gsum += red[i];
    int cnt = 0;
    for (int t = 0; t < TT; ++t) cnt += (mrow[t] > 0) ? 1 : 0;
    const int l0    = lrow[0];
    const int llast = lrow[cnt - 1];
    const float gold = startv[l0] + gsum + endv[llast];
    ll_ws[b] = gold - s_logZ;
  }
}

// ---------------------------------------------------------------------------
// Kernel 3: loss = -sum_b ll[b], fixed-order (deterministic).
// ---------------------------------------------------------------------------
__global__ void crf_finalize(const float* __restrict__ ll, float* __restrict__ loss) {
  if (threadIdx.x == 0 && blockIdx.x == 0) {
    float s = 0.0f;
    for (int b = 0; b < BB; ++b) s += ll[b];
    *loss = -s;
  }
}

extern "C" void kernel_launch(void* const* d_in, const int* in_sizes, int n_in,
                              void* d_out, int out_size, void* d_ws, size_t ws_size,
                              hipStream_t stream) {
  const float* inputs = (const float*)d_in[0];  // [B,T,D]
  const int*   mask   = (const int*)  d_in[1];  // [B,T]
  const int*   labels = (const int*)  d_in[2];  // [B,T]
  const float* W      = (const float*)d_in[3];  // [D,K]
  const float* bias   = (const float*)d_in[4];  // [K]
  const float* trans  = (const float*)d_in[5];  // [K,K]
  const float* startv = (const float*)d_in[6];  // [K]
  const float* endv   = (const float*)d_in[7];  // [K]

  float* scores = (float*)d_out;                          // [B,T,K]
  float* tags   = scores + (size_t)BB * TT * KK;          // [B,T] as float
  float* loss   = tags + (size_t)BB * TT;                 // [1]
  float* ll_ws  = (float*)d_ws;                           // [B]

  // 1) emissions GEMM via f32 WMMA: 2048 M-tiles x 4 N-tiles, 4 waves/block
  crf_emissions_gemm<<<dim3(MROWS / 16), dim3(128), 0, stream>>>(
      inputs, W, bias, mask, scores);

  // 2) per-batch Viterbi + partition + gold score
  crf_dp<<<dim3(BB), dim3(64), 0, stream>>>(
      scores, mask, labels, trans, startv, endv, tags, ll_ws);

  // 3) deterministic loss reduction
  crf_finalize<<<dim3(1), dim3(32), 0, stream>>>(ll_ws, loss);
}